// FullPairwise_38250978738342
// MI455X (gfx1250) — compile-verified
//
#include <hip/hip_runtime.h>
#include <hip/hip_bf16.h>

typedef __attribute__((ext_vector_type(2))) float v2f;
typedef __attribute__((ext_vector_type(8))) float v8f;

#define CUTOFF2 (5.2f * 5.2f)

// One wave (32 lanes) computes one 16x16 tile-pair (ti, tj) of the pairwise
// distance matrix for molecule `mol` with a single V_WMMA_F32_16X16X4_F32.
// The norm trick uses the spare K=3 slot so no cross-lane shuffles are needed:
//   A row m    = [-2x, -2y, -2z, |xi|^2]
//   B column n = [ x,   y,   z,  1    ]
//   G[m][n]    = |xi[m]|^2 - 2*xi[m].xj[n]
//   d2         = G[m][n] + |xj[n]|^2      (|xj|^2 is lane-local: n == lane&15)
__global__ __launch_bounds__(256) void pairwise_wmma_kernel(
    const int* __restrict__ species,
    const float* __restrict__ coords,
    float* __restrict__ out,
    int M, int N, long long P)
{
    const int wave = threadIdx.x >> 5;          // 8 waves per block
    const int lane = threadIdx.x & 31;          // wave32
    const int lo   = lane & 15;
    const int hi   = lane >> 4;

    const int tj  = blockIdx.x;                 // tile column
    const int ti  = blockIdx.y * 8 + wave;      // tile row (wave-uniform)
    const int mol = blockIdx.z;
    if (ti > tj) return;                        // wave-uniform: EXEC stays full

    const float* cm = coords  + (size_t)mol * N * 3;
    const int*   sm = species + (size_t)mol * N;

    // Lane `lo` owns row (ti*16+lo) of the A-tile and column (tj*16+lo) of B.
    const int gi = ti * 16 + lo;
    const int gj = tj * 16 + lo;

    float xix = cm[gi * 3 + 0], xiy = cm[gi * 3 + 1], xiz = cm[gi * 3 + 2];
    float xjx = cm[gj * 3 + 0], xjy = cm[gj * 3 + 1], xjz = cm[gj * 3 + 2];
    // Padding atoms (species == -1) -> NaN coords -> NaN d2 -> mask false.
    if (sm[gi] == -1) { xix = xiy = xiz = __builtin_nanf(""); }
    if (sm[gj] == -1) { xjx = xjy = xjz = __builtin_nanf(""); }

    const float ni = xix * xix + xiy * xiy + xiz * xiz;  // |row lo of tile i|^2
    const float nj = xjx * xjx + xjy * xjy + xjz * xjz;  // |col lo of tile j|^2

    // A 16x4 f32 layout: VGPR0 = K0 (lanes 0-15) / K2 (lanes 16-31),
    //                    VGPR1 = K1 / K3. B (4x16) mirrors by K-half.
    v2f a, b;
    a.x = hi ? (-2.0f * xiz) : (-2.0f * xix);
    a.y = hi ? ni            : (-2.0f * xiy);
    b.x = hi ? xjz           : xjx;
    b.y = hi ? 1.0f          : xjy;

    v8f c = {};
    // D = A x B + 0 : per-(m,n) value  |xi|^2 - 2*xi.xj
    v8f g = __builtin_amdgcn_wmma_f32_16x16x4_f32(
        /*neg_a=*/false, a, /*neg_b=*/false, b,
        /*c_mod=*/(short)0, c, /*reuse_a=*/false, /*reuse_b=*/false);

    // Output sections (concatenated flat, reference return order). All bases
    // are wave-uniform; per-element indexing is unsigned 32-bit so the backend
    // can use SADDR + 32-bit voffset global stores.
    float* __restrict__ oI1 = out + (long long)mol * 2 * P;          // idx i
    float* __restrict__ oI2 = oI1 + P;                                // idx j
    float* __restrict__ oM  = out + (long long)M * 2 * P + (long long)mol * P;
    float* __restrict__ oD  = out + (long long)M * 3 * P + (long long)mol * P;

    const int   molOff = mol * N;
    const int   gjc    = tj * 16 + lo;          // D layout: N = lane & 15
    const float fGj    = (float)(gjc + molOff); // loop-invariant per lane

    #pragma unroll
    for (int r = 0; r < 8; ++r) {
        const int gir = ti * 16 + (r + 8 * hi); // D layout: M = r + 8*(lane>=16)
        if (gjc > gir) {                        // strict upper triangle only
            const float d2  = g[r] + nj;
            const bool  msk = d2 <= CUTOFF2;    // NaN -> false
            // row-major triu(k=1) flat pair index; product < 2^23, int32 safe
            const unsigned p =
                (unsigned)(gir * (2 * N - gir - 1) / 2 + (gjc - gir - 1));
            oI1[p] = msk ? (float)(gir + molOff) : -1.0f;
            oI2[p] = msk ? fGj : -1.0f;
            oM[p]  = msk ? 1.0f : 0.0f;
            oD[p]  = d2;
        }
    }

    // shift_values = {0,0,0}, written once by the first wave of block (0,0,0).
    if (mol == 0 && ti == 0 && tj == 0 && lane < 3) {
        out[(long long)M * 4 * P + lane] = 0.0f;
    }
}

extern "C" void kernel_launch(void* const* d_in, const int* in_sizes, int n_in,
                              void* d_out, int out_size, void* d_ws, size_t ws_size,
                              hipStream_t stream) {
    const int*   species = (const int*)d_in[0];
    const float* coords  = (const float*)d_in[1];
    float*       out     = (float*)d_out;

    const int N = 2048;                         // atoms per molecule (reference)
    const int M = in_sizes[0] / N;              // molecules (=4)
    const long long P = (long long)N * (N - 1) / 2;
    const int T = N / 16;                       // 128 tiles per dimension

    dim3 grid(T, T / 8, M);                     // tj, ti/8, mol
    dim3 block(256);                            // 8 wave32 per block
    hipLaunchKernelGGL(pairwise_wmma_kernel, grid, block, 0, stream,
                       species, coords, out, M, N, P);
}